// GNNEncoder_21251498180833
// MI455X (gfx1250) — compile-verified
//
#include <hip/hip_runtime.h>

typedef float v2f __attribute__((ext_vector_type(2)));
typedef float v8f __attribute__((ext_vector_type(8)));

#define BN_EPS 1e-5f

// ---------------- utility ----------------
__global__ void k_fill_zero(float* __restrict__ p, long long n) {
  long long t = (long long)blockIdx.x * blockDim.x + threadIdx.x;
  if (t < n) p[t] = 0.0f;
}

// pad x [N x 3] -> xpad [N x 4] (zero 4th column) so layer-0 GEMM needs no K bounds checks
__global__ void k_pad_x(const float* __restrict__ x, float* __restrict__ xp, int n) {
  int t = blockIdx.x * blockDim.x + threadIdx.x;
  if (t >= n * 4) return;
  int i = t >> 2, j = t & 3;
  xp[t] = (j < 3) ? x[i * 3 + j] : 0.0f;
}

// deg[col[e]] += ew[e]  (self-loop weight 1 added in k_dinv)
__global__ void k_deg_edges(const int* __restrict__ ei, const float* __restrict__ ew,
                            float* __restrict__ deg, int E) {
  int e = blockIdx.x * blockDim.x + threadIdx.x;
  if (e < E) atomicAdd(&deg[ei[E + e]], ew[e]);
}

// dinv[i] = rsqrt(deg[i] + 1)
__global__ void k_dinv(float* __restrict__ deg, int n) {
  int i = blockIdx.x * blockDim.x + threadIdx.x;
  if (i < n) deg[i] = rsqrtf(deg[i] + 1.0f);
}

// norm[e] = dinv[row] * ew * dinv[col]   (layer-invariant)
__global__ void k_edge_norm(const int* __restrict__ ei, const float* __restrict__ ew,
                            const float* __restrict__ dinv, float* __restrict__ norm, int E) {
  int e = blockIdx.x * blockDim.x + threadIdx.x;
  if (e < E) norm[e] = dinv[ei[e]] * ew[e] * dinv[ei[E + e]];
}

// ---------------- WMMA fp32 GEMM: C[n x DOUT] = A[n x DIN] * W[DIN x DOUT] ----------------
// compile-time dims; DIN % 4 == 0, DOUT % 16 == 0, n % 16 == 0.
// One wave per 16-row block; it computes all DOUT/16 column tiles, reusing the A fragment.
template <int DIN, int DOUT>
__global__ void k_gemm_wmma(const float* __restrict__ A, const float* __restrict__ W,
                            float* __restrict__ C, int n) {
  constexpr int NT = DOUT / 16;
  int lane = threadIdx.x & 31;
  int wave = blockIdx.x * (blockDim.x >> 5) + (threadIdx.x >> 5);
  int tiles_m = n >> 4;
  if (wave >= tiles_m) return;                      // wave-uniform exit; EXEC stays full
  int half = lane >> 4;                             // 0: K=0,1   1: K=2,3
  int l15  = lane & 15;
  int m    = (wave << 4) + l15;                     // A row for this lane

  v8f acc[NT];
#pragma unroll
  for (int t = 0; t < NT; ++t) acc[t] = (v8f){};

#pragma unroll
  for (int k = 0; k < DIN; k += 4) {
    int kb = k + half * 2;                          // even -> 8B-aligned v2f load
    v2f a = *(const v2f*)&A[(long long)m * DIN + kb];
#pragma unroll
    for (int t = 0; t < NT; ++t) {
      int ncol = t * 16 + l15;
      v2f b;
      b.x = W[(long long)kb * DOUT + ncol];
      b.y = W[(long long)(kb + 1) * DOUT + ncol];
      acc[t] = __builtin_amdgcn_wmma_f32_16x16x4_f32(
          false, a, false, b, (short)0, acc[t], false, false);
    }
  }

  int mbase = (wave << 4) + half * 8;               // C layout: vgpr i -> row i (+8 high half)
#pragma unroll
  for (int t = 0; t < NT; ++t) {
    int ncol = t * 16 + l15;
#pragma unroll
    for (int i = 0; i < 8; ++i)
      C[(long long)(mbase + i) * DOUT + ncol] = acc[t][i];
  }
}

// out[i,f] = hW[i,f]*dinv[i]^2 + b[f]   (self-loop + bias as scatter initializer)
__global__ void k_init_out(const float* __restrict__ hW, const float* __restrict__ dinv,
                           const float* __restrict__ bias, float* __restrict__ out,
                           long long nd, int sh) {
  long long t = (long long)blockIdx.x * blockDim.x + threadIdx.x;
  if (t >= nd) return;
  int i = (int)(t >> sh);
  int f = (int)(t & ((1 << sh) - 1));
  float dv = dinv[i];
  out[t] = hW[t] * dv * dv + bias[f];
}

// edge scatter: out[col,f] += hW[row,f] * norm[e]; one thread per (edge, feature)
__global__ void k_scatter(const float* __restrict__ hW, const int* __restrict__ ei,
                          const float* __restrict__ norm, float* __restrict__ out,
                          int E, int sh) {
  long long t = (long long)blockIdx.x * blockDim.x + threadIdx.x;
  long long total = (long long)E << sh;
  if (t >= total) return;
  int e = (int)(t >> sh);
  int f = (int)(t & ((1 << sh) - 1));
  int s = ei[e];
  int d = ei[E + e];
  int dout = 1 << sh;
  atomicAdd(&out[(long long)d * dout + f], hW[(long long)s * dout + f] * norm[e]);
}

// per-feature sum / sumsq; total threads must be a multiple of dout
__global__ void k_bn_stats(const float* __restrict__ h, float* __restrict__ stats,
                           int n, int sh) {
  int tid   = blockIdx.x * blockDim.x + threadIdx.x;
  int total = gridDim.x * blockDim.x;
  int dout  = 1 << sh;
  int f     = tid & (dout - 1);
  int r0    = tid >> sh;
  int rows  = total >> sh;
  float s = 0.0f, s2 = 0.0f;
  for (int i = r0; i < n; i += rows) {
    float v = h[(long long)i * dout + f];
    s += v; s2 += v * v;
  }
  atomicAdd(&stats[f], s);
  atomicAdd(&stats[dout + f], s2);
}

// fold mean/var/gamma/beta into fused scale/shift
__global__ void k_bn_finalize(float* __restrict__ stats, const float* __restrict__ gamma,
                              const float* __restrict__ beta, int n, int dout) {
  int f = threadIdx.x;
  if (f >= dout) return;
  float inv_n = 1.0f / (float)n;
  float mu   = stats[f] * inv_n;
  float var  = stats[dout + f] * inv_n - mu * mu;
  float rstd = rsqrtf(var + BN_EPS);
  float sc   = rstd * gamma[f];
  stats[2 * dout + f] = sc;
  stats[3 * dout + f] = beta[f] - mu * sc;
}

// h_next = relu(h*scale + shift)
__global__ void k_bn_apply_relu(const float* __restrict__ h, const float* __restrict__ scale,
                                const float* __restrict__ shift, float* __restrict__ out,
                                long long nd, int sh) {
  long long t = (long long)blockIdx.x * blockDim.x + threadIdx.x;
  if (t >= nd) return;
  int f = (int)(t & ((1 << sh) - 1));
  float v = h[t] * scale[f] + shift[f];
  out[t] = fmaxf(v, 0.0f);
}

// global mean pool accumulation: one thread per (node, feature); f==0 also counts
__global__ void k_pool_accum(const float* __restrict__ ne, const int* __restrict__ batch,
                             float* __restrict__ psum, float* __restrict__ pcnt, long long nf) {
  long long t = (long long)blockIdx.x * blockDim.x + threadIdx.x;
  if (t >= nf) return;
  int i = (int)(t >> 4);
  int f = (int)(t & 15);
  int g = batch[i];
  atomicAdd(&psum[g * 16 + f], ne[t]);
  if (f == 0) atomicAdd(&pcnt[g], 1.0f);
}

__global__ void k_pool_final(const float* __restrict__ psum, const float* __restrict__ pcnt,
                             float* __restrict__ out) {
  int t = blockIdx.x * blockDim.x + threadIdx.x;
  if (t >= 64 * 16) return;
  int g = t >> 4;
  out[t] = psum[t] / fmaxf(pcnt[g], 1.0f);
}

// ---------------- launcher ----------------
extern "C" void kernel_launch(void* const* d_in, const int* in_sizes, int n_in,
                              void* d_out, int out_size, void* d_ws, size_t ws_size,
                              hipStream_t stream) {
  const float* x     = (const float*)d_in[0];
  const float* ew    = (const float*)d_in[1];
  const int*   ei    = (const int*)d_in[2];
  const int*   batch = (const int*)d_in[3];
  const float* Wm[3]    = {(const float*)d_in[4],  (const float*)d_in[8],  (const float*)d_in[12]};
  const float* bias[3]  = {(const float*)d_in[5],  (const float*)d_in[9],  (const float*)d_in[13]};
  const float* gamma[3] = {(const float*)d_in[6],  (const float*)d_in[10], (const float*)d_in[14]};
  const float* beta[3]  = {(const float*)d_in[7],  (const float*)d_in[11], (const float*)d_in[15]};

  const int N = in_sizes[0] / 3;
  const int E = in_sizes[1];
  const int dims[4] = {3, 64, 32, 16};

  float* ws    = (float*)d_ws;
  float* hW    = ws;                       // N*64
  float* hout  = hW   + (size_t)N * 64;    // N*64
  float* hbuf  = hout + (size_t)N * 64;    // N*64
  float* dinv  = hbuf + (size_t)N * 64;    // N
  float* norm  = dinv + N;                 // E
  float* stats = norm + E;                 // 4*64 (sum, sumsq, scale, shift)
  float* psum  = stats + 256;              // 64*16
  float* pcnt  = psum + 64 * 16;           // 64
  float* xpad  = pcnt + 64;                // N*4

  float* outF    = (float*)d_out;          // [64*16 graph emb][N*16 node emb]
  float* nodeEmb = outF + 64 * 16;

  const int B = 256;

  // degrees -> dinv -> per-edge norms; pad x to K=4
  k_fill_zero<<<(N + B - 1) / B, B, 0, stream>>>(dinv, N);
  k_deg_edges<<<(E + B - 1) / B, B, 0, stream>>>(ei, ew, dinv, E);
  k_dinv<<<(N + B - 1) / B, B, 0, stream>>>(dinv, N);
  k_edge_norm<<<(E + B - 1) / B, B, 0, stream>>>(ei, ew, dinv, norm, E);
  k_pad_x<<<(N * 4 + B - 1) / B, B, 0, stream>>>(x, xpad, N);

  const float* hin = xpad;
  int gemm_blocks = ((N >> 4) + 3) / 4;    // one wave per 16-row block, 4 waves/block
  for (int l = 0; l < 3; ++l) {
    const int dout = dims[l + 1];
    const int sh = (dout == 64) ? 6 : (dout == 32) ? 5 : 4;
    const long long nd = (long long)N * dout;

    // dense transform via WMMA (compile-time shapes)
    if (l == 0)      k_gemm_wmma<4, 64> <<<gemm_blocks, 128, 0, stream>>>(hin, Wm[l], hW, N);
    else if (l == 1) k_gemm_wmma<64, 32><<<gemm_blocks, 128, 0, stream>>>(hin, Wm[l], hW, N);
    else             k_gemm_wmma<32, 16><<<gemm_blocks, 128, 0, stream>>>(hin, Wm[l], hW, N);

    // self-loop + bias init, then edge scatter-add
    k_init_out<<<(int)((nd + B - 1) / B), B, 0, stream>>>(hW, dinv, bias[l], hout, nd, sh);
    long long edd = (long long)E << sh;
    k_scatter<<<(int)((edd + B - 1) / B), B, 0, stream>>>(hW, ei, norm, hout, E, sh);

    // batchnorm (training-mode stats) + relu
    k_fill_zero<<<1, B, 0, stream>>>(stats, 2 * dout);
    k_bn_stats<<<256, B, 0, stream>>>(hout, stats, N, sh);
    k_bn_finalize<<<1, 64, 0, stream>>>(stats, gamma[l], beta[l], N, dout);
    float* hdst = (l == 2) ? nodeEmb : hbuf;
    k_bn_apply_relu<<<(int)((nd + B - 1) / B), B, 0, stream>>>(hout, stats + 2 * dout,
                                                               stats + 3 * dout, hdst, nd, sh);
    hin = hbuf;
  }

  // global mean pool over sorted batch ids
  k_fill_zero<<<(64 * 16 + 64 + B - 1) / B, B, 0, stream>>>(psum, 64 * 16 + 64);
  long long nf = (long long)N * 16;
  k_pool_accum<<<(int)((nf + B - 1) / B), B, 0, stream>>>(nodeEmb, batch, psum, pcnt, nf);
  k_pool_final<<<4, B, 0, stream>>>(psum, pcnt, outF);
}